// NeuralODE_53635551593026
// MI455X (gfx1250) — compile-verified
//
#include <hip/hip_runtime.h>
#include <math.h>

// ---------------------------------------------------------------------------
// NeuralODE on MI455X (gfx1250).
//
// 255 intervals x 3 RK4 substeps x 4 MLP evals = 3060 chained 3-layer MLPs on
// a 128x512 state (~4.9 TFLOP total).  Weights (24 MB fp32) are L2-resident
// (192 MB); the problem is sequential-latency bound, not HBM bound, so we use
// the full-precision V_WMMA_F32_16X16X4_F32 matrix path (matches reference
// fp32 numerics over thousands of chained layers).
//
// This revision:
//  * GEMM shapes are template constants -> all address math is shifts/adds,
//    pointer-increment K-loop (kills the v_mul_u64 / divide seen in round 1).
//  * Weights are repacked ONCE per launch into WMMA-native per-lane float4
//    groups, so the B operand is a single coalesced global_load_b128 per
//    k-step (was 4 strided b32 loads).  A operand stays one global_load_b64.
// ---------------------------------------------------------------------------

typedef __attribute__((ext_vector_type(2))) float v2f;
typedef __attribute__((ext_vector_type(4))) float v4f;
typedef __attribute__((ext_vector_type(8))) float v8f;

#define EP_TANH 0
#define EP_KACC 1
#define EP_LAST 2

// ---------------------------------------------------------------------------
// Weight repack: W[K][N] row-major -> packed per-lane float4 groups.
// Packed element (nt, kt, lane) holds the 4 B-values lane needs for the
// 16x32 wave tile at n0 = 32*nt, k-group kt:
//   { W[ka][n0+l15], W[ka+1][n0+l15], W[ka][n0+16+l15], W[ka+1][n0+16+l15] }
// with ka = 4*kt + 2*(lane>>4), l15 = lane&15.  Layout is nt-major, then kt,
// then lane -> the GEMM K-loop streams 512 contiguous bytes per wave per kt.
// ---------------------------------------------------------------------------
template <int K, int N>
__global__ __launch_bounds__(256) void repack_w(const float* __restrict__ W,
                                                float* __restrict__ Wp)
{
  constexpr int KT = K / 4;
  const int idx = blockIdx.x * 256 + threadIdx.x;  // [0, (N/32)*KT*32)
  const int lane = idx & 31;
  const int t = idx >> 5;         // nt * KT + kt
  const int kt = t & (KT - 1);    // KT is a power of two
  const int nt = t / KT;
  const int half = lane >> 4;
  const int l15 = lane & 15;
  const int ka = 4 * kt + 2 * half;
  const float* w = W + (size_t)ka * N + nt * 32 + l15;
  v4f v;
  v.x = w[0];
  v.y = w[N];
  v.z = w[16];
  v.w = w[N + 16];
  ((v4f*)Wp)[idx] = v;
}

// ---------------------------------------------------------------------------
// Fused GEMM:  Out = epilogue( (A + alpha*Kprev) @ W + bias )
// Wave tile 16(M) x 32(N): two 16x16 f32 WMMA accumulators sharing one A
// fetch.  256 threads = 8 waves per workgroup, exact grid fill, no
// divergence before WMMA (EXEC all ones as the ISA requires).
// ---------------------------------------------------------------------------
template <int K, int N, bool FUSE_IN, int EP>
__global__ __launch_bounds__(256) void gemm_f32_wmma(
    const float* __restrict__ A,      // [128 x K]
    const float* __restrict__ Kprev,  // [128 x K] (FUSE_IN only)
    float alpha,
    const float* __restrict__ Wp,     // packed weights (repack_w layout)
    const float* __restrict__ bias,   // [N]
    float* __restrict__ Out,          // EP_TANH: h-store ; EP_KACC: k-store
    float* __restrict__ Kacc,         // EP_KACC: update ; EP_LAST: read
    float kw, int kaccInit,
    const float* __restrict__ Ybase,  // EP_LAST
    float* __restrict__ Ynew,         // EP_LAST (may alias Ybase)
    float* __restrict__ OutRow,       // EP_LAST: optional d_out row (or null)
    long long outStride, float dt6)
{
  constexpr int KT = K / 4;
  constexpr int NT = N / 32;

  const int lane = threadIdx.x & 31;
  const int wave = threadIdx.x >> 5;
  const int wtile = blockIdx.x * 8 + wave;  // global 16x32 wave-tile id
  const int mt = wtile / NT;                // power-of-two -> shift
  const int nt = wtile - mt * NT;
  const int m0 = mt << 4;
  const int n0 = nt << 5;

  const int half = lane >> 4;  // 0: K+{0,1}, rows m0..m0+7 ; 1: K+{2,3}, rows m0+8..
  const int l15 = lane & 15;

  // A operand (16x4 f32): lanes 0-15 = rows, VGPR0/1 = K+2h, K+2h+1.
  const float* Ap = A + (size_t)(m0 + l15) * K + 2 * half;
  const float* Kp = FUSE_IN ? (Kprev + (size_t)(m0 + l15) * K + 2 * half) : nullptr;
  // Packed B stream for this n-tile: one float4 (b128) per lane per k-group.
  const v4f* Bp = (const v4f*)Wp + (size_t)nt * KT * 32 + lane;

  v8f acc0 = {};
  v8f acc1 = {};

#pragma unroll 4
  for (int kt = 0; kt < KT; ++kt) {
    v2f a;
    if constexpr (FUSE_IN) {
      a.x = fmaf(alpha, Kp[0], Ap[0]);
      a.y = fmaf(alpha, Kp[1], Ap[1]);
      Kp += 4;
    } else {
      a = *(const v2f*)Ap;  // global_load_b64
    }
    Ap += 4;
    const v4f b = *Bp;  // global_load_b128, coalesced 512B/wave
    Bp += 32;
    v2f b0, b1;
    b0.x = b.x; b0.y = b.y;   // rows ka, ka+1 @ cols n0+l15
    b1.x = b.z; b1.y = b.w;   // rows ka, ka+1 @ cols n0+16+l15
    acc0 = __builtin_amdgcn_wmma_f32_16x16x4_f32(false, a, false, b0, (short)0,
                                                 acc0, false, false);
    acc1 = __builtin_amdgcn_wmma_f32_16x16x4_f32(false, a, false, b1, (short)0,
                                                 acc1, false, false);
  }

  // C/D layout: VGPR r -> row m0 + r + 8*half, col n0 + l15 (+16 for tile 1).
  const int c0 = n0 + l15;
  const int c1 = c0 + 16;
  const float bv0 = bias[c0];
  const float bv1 = bias[c1];

#pragma unroll
  for (int r = 0; r < 8; ++r) {
    const int row = m0 + r + 8 * half;
    const size_t i0 = (size_t)row * N + c0;
    const size_t i1 = (size_t)row * N + c1;
    const float v0 = acc0[r] + bv0;
    const float v1 = acc1[r] + bv1;
    if constexpr (EP == EP_TANH) {
      Out[i0] = tanhf(v0);
      Out[i1] = tanhf(v1);
    } else if constexpr (EP == EP_KACC) {
      Out[i0] = v0;
      Out[i1] = v1;
      if (kaccInit) { Kacc[i0] = kw * v0;  Kacc[i1] = kw * v1; }
      else          { Kacc[i0] += kw * v0; Kacc[i1] += kw * v1; }
    } else {  // EP_LAST: y += dt/6 * (kacc + k4), optional output-row store
      const float y0 = Ybase[i0] + dt6 * (Kacc[i0] + v0);
      const float y1 = Ybase[i1] + dt6 * (Kacc[i1] + v1);
      Ynew[i0] = y0;
      Ynew[i1] = y1;
      if (OutRow) {
        OutRow[(long long)row * outStride + c0] = y0;
        OutRow[(long long)row * outStride + c1] = y1;
      }
    }
  }
}

// y0 = x[:,0,:] ; also emit output timestep 0.
__global__ __launch_bounds__(256) void init_state(const float* __restrict__ x,
                                                  float* __restrict__ Y,
                                                  float* __restrict__ Out,
                                                  int TC, int C)
{
  const int i = blockIdx.x * 256 + threadIdx.x;  // [0, B*C)
  const int b = i / C;
  const int c = i - b * C;
  const float v = x[(size_t)b * TC + c];
  Y[i] = v;
  Out[(size_t)b * TC + c] = v;
}

extern "C" void kernel_launch(void* const* d_in, const int* in_sizes, int n_in,
                              void* d_out, int out_size, void* d_ws, size_t ws_size,
                              hipStream_t stream)
{
  const float* x  = (const float*)d_in[0];
  const float* W1 = (const float*)d_in[1];
  const float* b1 = (const float*)d_in[2];
  const float* W2 = (const float*)d_in[3];
  const float* b2 = (const float*)d_in[4];
  const float* W3 = (const float*)d_in[5];
  const float* b3 = (const float*)d_in[6];
  float* out = (float*)d_out;

  constexpr int B = 128, T = 256, C = 512, H = 2048;
  constexpr int N_SUB = 3;  // matches reference derivation
  const float DT  = (float)((1.0 / (double)(T - 1)) / (double)N_SUB);
  const float dt6 = DT / 6.0f;

  // Workspace: state/scratch (~2.75 MB) + packed weights (24 MB).
  float* Y    = (float*)d_ws;
  float* Kb   = Y    + (size_t)B * C;
  float* Kacc = Kb   + (size_t)B * C;
  float* H1   = Kacc + (size_t)B * C;
  float* H2   = H1   + (size_t)B * H;
  float* W1p  = H2   + (size_t)B * H;
  float* W2p  = W1p  + (size_t)C * H;
  float* W3p  = W2p  + (size_t)H * H;

  const dim3 blk(256);
  const int gH = 64;  // (128/16)*(2048/32) wave tiles / 8 waves per WG
  const int gC = 16;  // (128/16)*(512/32)  wave tiles / 8 waves per WG

  // One-time (per replay) weight repack into WMMA-native layout.
  repack_w<C, H><<<(C * H) / 1024, blk, 0, stream>>>(W1, W1p);
  repack_w<H, H><<<(H * H) / 1024, blk, 0, stream>>>(W2, W2p);
  repack_w<H, C><<<(H * C) / 1024, blk, 0, stream>>>(W3, W3p);

  init_state<<<(B * C) / 256, blk, 0, stream>>>(x, Y, out, T * C, C);

  const long long ostride = (long long)T * C;

  for (int t = 1; t < T; ++t) {
    for (int s = 0; s < N_SUB; ++s) {
      const bool last = (s == N_SUB - 1);
      float* orow = last ? (out + (size_t)t * C) : nullptr;

      // ---- eval 1: k1 = mlp(y); kacc = k1
      gemm_f32_wmma<C, H, false, EP_TANH><<<gH, blk, 0, stream>>>(
          Y, nullptr, 0.f, W1p, b1, H1, nullptr, 0.f, 0,
          nullptr, nullptr, nullptr, 0, 0.f);
      gemm_f32_wmma<H, H, false, EP_TANH><<<gH, blk, 0, stream>>>(
          H1, nullptr, 0.f, W2p, b2, H2, nullptr, 0.f, 0,
          nullptr, nullptr, nullptr, 0, 0.f);
      gemm_f32_wmma<H, C, false, EP_KACC><<<gC, blk, 0, stream>>>(
          H2, nullptr, 0.f, W3p, b3, Kb, Kacc, 1.f, 1,
          nullptr, nullptr, nullptr, 0, 0.f);

      // ---- eval 2: k2 = mlp(y + dt/2 * k1); kacc += 2*k2
      gemm_f32_wmma<C, H, true, EP_TANH><<<gH, blk, 0, stream>>>(
          Y, Kb, 0.5f * DT, W1p, b1, H1, nullptr, 0.f, 0,
          nullptr, nullptr, nullptr, 0, 0.f);
      gemm_f32_wmma<H, H, false, EP_TANH><<<gH, blk, 0, stream>>>(
          H1, nullptr, 0.f, W2p, b2, H2, nullptr, 0.f, 0,
          nullptr, nullptr, nullptr, 0, 0.f);
      gemm_f32_wmma<H, C, false, EP_KACC><<<gC, blk, 0, stream>>>(
          H2, nullptr, 0.f, W3p, b3, Kb, Kacc, 2.f, 0,
          nullptr, nullptr, nullptr, 0, 0.f);

      // ---- eval 3: k3 = mlp(y + dt/2 * k2); kacc += 2*k3
      gemm_f32_wmma<C, H, true, EP_TANH><<<gH, blk, 0, stream>>>(
          Y, Kb, 0.5f * DT, W1p, b1, H1, nullptr, 0.f, 0,
          nullptr, nullptr, nullptr, 0, 0.f);
      gemm_f32_wmma<H, H, false, EP_TANH><<<gH, blk, 0, stream>>>(
          H1, nullptr, 0.f, W2p, b2, H2, nullptr, 0.f, 0,
          nullptr, nullptr, nullptr, 0, 0.f);
      gemm_f32_wmma<H, C, false, EP_KACC><<<gC, blk, 0, stream>>>(
          H2, nullptr, 0.f, W3p, b3, Kb, Kacc, 2.f, 0,
          nullptr, nullptr, nullptr, 0, 0.f);

      // ---- eval 4: k4 = mlp(y + dt * k3); y += dt/6 * (kacc + k4)
      gemm_f32_wmma<C, H, true, EP_TANH><<<gH, blk, 0, stream>>>(
          Y, Kb, DT, W1p, b1, H1, nullptr, 0.f, 0,
          nullptr, nullptr, nullptr, 0, 0.f);
      gemm_f32_wmma<H, H, false, EP_TANH><<<gH, blk, 0, stream>>>(
          H1, nullptr, 0.f, W2p, b2, H2, nullptr, 0.f, 0,
          nullptr, nullptr, nullptr, 0, 0.f);
      gemm_f32_wmma<H, C, false, EP_LAST><<<gC, blk, 0, stream>>>(
          H2, nullptr, 0.f, W3p, b3, Kb, Kacc, 0.f, 0,
          Y, Y, orow, ostride, dt6);
    }
  }
}